// ViterbiDecoder_9990093930830
// MI455X (gfx1250) — compile-verified
//
#include <hip/hip_runtime.h>
#include <stdint.h>

// Smoothed affine-gap Viterbi/NW forward pass, anti-diagonal wavefront.
// N = M = 1024, one persistent 1024-thread workgroup (32 wave32 waves, 1 WGP).
// Thread tid owns row i = tid+1. Diagonal state lives in 3 rotated LDS
// buffers. Theta anti-diagonals (a strided 2D tile: 1024 rows x 12B, row
// stride 12276B) are staged into LDS by the CDNA5 Tensor Data Mover
// (tensor_load_to_lds, TENSORcnt), one descriptor per diagonal issued by
// wave 0, depth-1 pipelined and triple-buffered. Falls back to per-lane
// GLOBAL_LOAD_ASYNC_TO_LDS_B32 (ASYNCcnt) if the TDM builtin is absent.

#define N_LEN 1024
#define M_LEN 1024
#define KMAX  (N_LEN + M_LEN)        // last diagonal index (V[N,M] lives here)
#define TH_ELEMS (N_LEN * M_LEN * 3) // 3145728 floats in theta
#define ROW_STRIDE ((M_LEN - 1) * 3) // 3069 elements between diagonal rows

#if __has_builtin(__builtin_amdgcn_tensor_load_to_lds)
#define USE_TDM 1
#else
#define USE_TDM 0
#endif

__device__ __forceinline__ float lse2f(float a, float b) {
    float mx = fmaxf(a, b);
    return mx + __logf(__expf(a - mx) + __expf(b - mx));
}

__device__ __forceinline__ float lse3f(float a, float b, float c) {
    float mx = fmaxf(fmaxf(a, b), c);
    return mx + __logf(__expf(a - mx) + __expf(b - mx) + __expf(c - mx));
}

#if USE_TDM
// Issue one TDM descriptor: gather the theta anti-diagonal kk into LDS.
// 2D view: element = f32 (data_size=2), tile 3 x 1024, tensor_dim0_stride =
// 3069 elements. Rows whose fabricated address would leave the allocation are
// cut off via tensor_dim1 and zero-filled by the TDM (always masked cells).
__device__ __forceinline__ void tdm_prefetch_diag(const float* theta, int kk,
                                                  float* lds_dst) {
    typedef unsigned int v4u __attribute__((ext_vector_type(4)));
    typedef int          v8i __attribute__((ext_vector_type(8)));
    typedef int          v4i __attribute__((ext_vector_type(4)));

    unsigned long long ga =
        (unsigned long long)(uintptr_t)theta + 12ull * (unsigned)(kk - 2);
    unsigned lds = (unsigned)(uintptr_t)lds_dst;

    // #rows y with 3*(kk-2) + 3069*y + 3 <= TH_ELEMS (fully in-bounds rows).
    int dim1 = ((TH_ELEMS - 3) - 3 * (kk - 2)) / ROW_STRIDE + 1;
    if (dim1 > N_LEN) dim1 = N_LEN;

    v4u g0;
    g0.x = 1u;                                   // count=1, user descriptor
    g0.y = lds;                                  // lds_addr
    g0.z = (unsigned)(ga & 0xFFFFFFFFu);         // global_addr[31:0]
    g0.w = (unsigned)((ga >> 32) & 0x01FFFFFFu)  // global_addr[56:32]
         | 0x80000000u;                          // type = 2 ("image")

    v8i g1;
    g1[0] = 0x00020000;        // workgroup_mask=0, data_size=2 (4B)
    g1[1] = 3 << 16;           // tensor_dim0 = 3 (bits 79:48, low part)
    g1[2] = dim1 << 16;        // tensor_dim0 hi = 0 | tensor_dim1 low16
    g1[3] = 3 << 16;           // tensor_dim1 hi = 0 | tile_dim0 = 3
    g1[4] = N_LEN;             // tile_dim1 = 1024, tile_dim2 = 0 (2D)
    g1[5] = ROW_STRIDE;        // tensor_dim0_stride = 3069 elements
    g1[6] = 0;                 // stride0 hi16 = 0, tensor_dim1_stride lo = 0
    g1[7] = 0;

    v4i gz = {0, 0, 0, 0};
#if __clang_major__ >= 23
    v8i gz8 = {0, 0, 0, 0, 0, 0, 0, 0};
    __builtin_amdgcn_tensor_load_to_lds(g0, g1, gz, gz, gz8, 0);
#else
    __builtin_amdgcn_tensor_load_to_lds(g0, g1, gz, gz, 0);
#endif
}

__device__ __forceinline__ void wait_tensorcnt_le1() {
#if __has_builtin(__builtin_amdgcn_s_wait_tensorcnt)
    __builtin_amdgcn_s_wait_tensorcnt((short)1);
#else
    asm volatile("s_wait_tensorcnt 0x1" ::: "memory");
#endif
}
#else  // !USE_TDM: proven per-lane async fallback -------------------------
__device__ __forceinline__ void async_g2l_b32(const float* g, unsigned ldsaddr) {
    asm volatile("global_load_async_to_lds_b32 %0, %1, off"
                 :: "v"(ldsaddr), "v"(g)
                 : "memory");
}
__device__ __forceinline__ void wait_asynccnt_le3() {
    asm volatile("s_wait_asynccnt 0x3" ::: "memory");
}
#endif

__global__ __launch_bounds__(N_LEN)
void viterbi_wavefront_kernel(const float* __restrict__ theta,
                              const float* __restrict__ A,
                              float* __restrict__ out) {
    // Rotating diagonal buffers: entry i of a buffer holds V[i, k-i][3].
    __shared__ float vbuf[3][(N_LEN + 1) * 3];
    // Rotating theta staging buffers, one anti-diagonal each (DMA-filled).
    __shared__ float tbuf[3][N_LEN * 3];

    const int tid = (int)threadIdx.x;   // 0..1023
    const int i   = tid + 1;            // row owned by this thread

    const float n = A[0];
    const float d = A[1];
    const float e = A[2];
    const float t = A[3];
    // y state reads vleft.m in BOTH branches -> collapses to vleft.m + Cy.
    const float Cy = lse2f(-d, -e);

    // Boundary condition: every diagonal entry starts at 0.0 (entry 0 of
    // each buffer is never rewritten, so rows/cols 0 stay 0 forever).
    {
        float* vb = (float*)vbuf;
        for (int idx = tid; idx < 3 * (N_LEN + 1) * 3; idx += N_LEN)
            vb[idx] = 0.0f;
    }

    // Rotated pointers (no per-iteration modulo arithmetic on the SALU).
    float* vA = vbuf[0];   // diag k-2
    float* vB = vbuf[1];   // diag k-1
    float* vC = vbuf[2];   // diag k   (write target)

    float* tcur   = tbuf[0];   // theta for diagonal k
    float* tnext  = tbuf[1];   // being filled for diagonal k+1
    float* tspare = tbuf[2];

#if !USE_TDM
    // Per-lane fallback prefetch: inactive cells CLAMPED to cell 0 so every
    // lane stays active and in-bounds.
    auto prefetch = [&](int kk, float* dst) {
        int j      = kk - i;
        int active = (j >= 1) && (j <= M_LEN);
        int cell   = active ? (tid * M_LEN + (j - 1)) : 0;
        const float* g = theta + 3u * (unsigned)cell;
        unsigned l = (unsigned)(uintptr_t)(dst + tid * 3);
        async_g2l_b32(g + 0, l + 0u);
        async_g2l_b32(g + 1, l + 4u);
        async_g2l_b32(g + 2, l + 8u);
    };
#endif

    // Warm the pipeline for the first diagonal (k = 2).
#if USE_TDM
    if (tid < 32) tdm_prefetch_diag(theta, 2, tcur);
#else
    prefetch(2, tcur);
#endif

    // vleft register: V[i, (k-1)-i]; diagonal k=1 is all zeros.
    float vm = 0.0f, vx = 0.0f, vy = 0.0f;

    for (int k = 2; k <= KMAX; ++k) {
        // Issue the DMA for diagonal k+1, then make diagonal k's data and the
        // previous V writes visible. k+1 == KMAX+1 is a harmless dummy.
#if USE_TDM
        if (tid < 32) {                 // wave 0 drives the TDM
            tdm_prefetch_diag(theta, k + 1, tnext);
            wait_tensorcnt_le1();       // in-order: DMA for diagonal k landed
        }
#else
        prefetch(k + 1, tnext);
        wait_asynccnt_le3();            // this wave's copies for k landed
#endif
        __syncthreads();

        // Neighbor reads (entry i-1 == tid); write buffer is distinct.
        const float* vU = vB + tid * 3;   // V[i-1, j]   (diag k-1)
        const float* vD = vA + tid * 3;   // V[i-1, j-1] (diag k-2)

        float d0 = vD[0], d1 = vD[1], d2 = vD[2];
        if (k == 2 && tid == 0) d0 = 2.0f * n;   // V[0,0,m] = 2n

        float m_val = lse3f(d0, d1, d2);
        float x_val = lse2f(vU[0] - d, vU[1] - e);
        float y_val = vm + Cy;

        const float* th = tcur + tid * 3;
        int  j   = k - i;
        bool act = (j >= 1) && (j <= M_LEN);
        float nm = act ? (m_val + th[0]) : 0.0f;
        float nx = act ? (x_val + th[1]) : 0.0f;
        float ny = act ? (y_val + th[2]) : 0.0f;

        float* w = vC + i * 3;
        w[0] = nm; w[1] = nx; w[2] = ny;
        vm = nm; vx = nx; vy = ny;

        // Rotate diagonal buffers and theta staging buffers.
        float* tmpv = vA; vA = vB; vB = vC; vC = tmpv;
        float* tmpt = tcur; tcur = tnext; tnext = tspare; tspare = tmpt;
    }

    // Thread owning row N computed V[N, M] on diagonal k = N + M; it still
    // holds the three states in registers. Apply the terminal correction.
    if (tid == N_LEN - 1) {
        const float ie    = 1e-10f;
        float eps   = (1.0f - t) * __expf(-e);
        float net   = (1.0f - n) / (1.0f - e - t);
        float delta = (1.0f - 2.0f * __expf(-d) * net) * net * (1.0f - t) * __expf(-d);
        float c     = __logf(1.0f - 2.0f * delta - t + ie)
                    - __logf(1.0f - eps - t + ie);
        out[0] = lse3f(vm, vx + c, vy + c);
    }
}

extern "C" void kernel_launch(void* const* d_in, const int* in_sizes, int n_in,
                              void* d_out, int out_size, void* d_ws, size_t ws_size,
                              hipStream_t stream) {
    (void)in_sizes; (void)n_in; (void)out_size; (void)d_ws; (void)ws_size;
    const float* theta = (const float*)d_in[0];   // [1024,1024,3] f32
    const float* A     = (const float*)d_in[1];   // [4] f32
    float* out = (float*)d_out;                   // scalar f32
    viterbi_wavefront_kernel<<<1, N_LEN, 0, stream>>>(theta, A, out);
}